// InvertedDotProductAttention_90855738179901
// MI455X (gfx1250) — compile-verified
//
#include <hip/hip_runtime.h>
#include <hip/hip_bf16.h>

typedef float v2f __attribute__((ext_vector_type(2)));
typedef float v4f __attribute__((ext_vector_type(4)));
typedef float v8f __attribute__((ext_vector_type(8)));
typedef int   v4i __attribute__((vector_size(4 * sizeof(int))));

#define AS1 __attribute__((address_space(1)))
#define AS3 __attribute__((address_space(3)))

#define BATCHES 64
#define QDIM 16
#define DDIM 64
#define KVDIM 16384
#define SPLIT 32
#define KEYS_PER_SPLIT (KVDIM / SPLIT)        /* 512  */
#define TILES_PER_SPLIT (KEYS_PER_SPLIT / 16) /* 32   */
#define KSTRIDE 68  /* 64 + 4 pad dwords: conflict-free b64 gathers   */
#define VSTRIDE 72  /* 64 + 8 pad dwords: conflict-free b32 row pairs */
#define ASTRIDE 18  /* 16 + 2 pad dwords */

#if __has_builtin(__builtin_amdgcn_global_load_async_to_lds_b128)
#define HAVE_ASYNC_LDS 1
#else
#define HAVE_ASYNC_LDS 0
#endif

#if __has_builtin(__builtin_amdgcn_s_wait_asynccnt)
#define WAIT_ASYNCCNT(n) __builtin_amdgcn_s_wait_asynccnt(n)
#else
#define WAIT_ASYNCCNT(n) asm volatile("s_wait_asynccnt %0" ::"n"(n) : "memory")
#endif

#if HAVE_ASYNC_LDS
// Stage a 16x64 f32 tile (4KB) global -> LDS with padded rows, 8 async b128 ops
__device__ __forceinline__ void stage_async(const float* __restrict__ g,
                                            float* l, int lane, int strideDwords) {
#pragma unroll
  for (int r = 0; r < 8; ++r) {
    const int f = r * 32 + lane;  // 16B unit index 0..255
    const int row = f >> 4;
    const int cg = f & 15;
    __builtin_amdgcn_global_load_async_to_lds_b128(
        (AS1 v4i*)(g + f * 4),
        (AS3 v4i*)(l + row * strideDwords + cg * 4), 0, 0);
  }
}
#endif

__global__ __launch_bounds__(32)
void inv_attn_partial(const float* __restrict__ Q,
                      const float* __restrict__ K,
                      const float* __restrict__ V,
                      float* __restrict__ wsO,
                      float* __restrict__ wsS) {
  __shared__ __align__(16) float kst[2][16 * KSTRIDE];
#if HAVE_ASYNC_LDS
  __shared__ __align__(16) float vst[2][16 * VSTRIDE];
#endif
  __shared__ __align__(16) float ast[16 * ASTRIDE];

  const int lane = threadIdx.x;   // 0..31 (wave32)
  const int half = lane >> 4;     // 0 or 1
  const int lm   = lane & 15;
  const int b = blockIdx.x / SPLIT;
  const int s = blockIdx.x % SPLIT;
  const int k_begin = s * KEYS_PER_SPLIT;

  const float* qb = Q + (size_t)b * (QDIM * DDIM);
  const float* kb = K + (size_t)b * KVDIM * DDIM + (size_t)k_begin * DDIM;
  const float* vb = V + (size_t)b * KVDIM * DDIM + (size_t)k_begin * DDIM;

  // Q in WMMA A-layout (16x4 f32 chunks), pre-scaled by 1/sqrt(D)=0.125
  v2f qa[16];
#pragma unroll
  for (int ci = 0; ci < 16; ++ci) {
    const int c = ci * 4;
    v2f t = *(const v2f*)(qb + lm * DDIM + c + 2 * half);
    qa[ci] = t * 0.125f;
  }

  v8f acc[4];
#pragma unroll
  for (int ni = 0; ni < 4; ++ni) acc[ni] = (v8f){0.f,0.f,0.f,0.f,0.f,0.f,0.f,0.f};
  float sp[8];
#pragma unroll
  for (int v = 0; v < 8; ++v) sp[v] = 0.f;

#if HAVE_ASYNC_LDS
  // Prime the pipeline: tile 0 -> buffer 0
  stage_async(kb, kst[0], lane, KSTRIDE);
  stage_async(vb, vst[0], lane, VSTRIDE);
#endif

  for (int t = 0; t < TILES_PER_SPLIT; ++t) {
    const int cur = t & 1;
    const float* ktile = kb + (size_t)t * 16 * DDIM;
    const float* vtile = vb + (size_t)t * 16 * DDIM;

#if HAVE_ASYNC_LDS
    if (t + 1 < TILES_PER_SPLIT) {
      stage_async(ktile + 16 * DDIM, kst[cur ^ 1], lane, KSTRIDE);
      stage_async(vtile + 16 * DDIM, vst[cur ^ 1], lane, VSTRIDE);
      WAIT_ASYNCCNT(16);  // oldest 16 (current tile) complete, next 16 in flight
    } else {
      WAIT_ASYNCCNT(0);
    }
    asm volatile("" ::: "memory");
    const float* kcur = kst[cur];
    const float* vcur = vst[cur];
#else
    // Synchronous fallback: stage K tile via b128 loads, V read direct from global
#pragma unroll
    for (int r = 0; r < 8; ++r) {
      const int f = r * 32 + lane;
      const int row = f >> 4;
      const int cg = f & 15;
      v4f kv = *(const v4f*)(ktile + f * 4);
      *(v4f*)&kst[0][row * KSTRIDE + cg * 4] = kv;
    }
    __syncthreads();
    const float* kcur = kst[0];
#endif

    // Logits tile L (16q x 16k) = Qs(16x64) @ Ktile^T, 16 x WMMA f32 16x16x4
    v8f L = (v8f){0.f,0.f,0.f,0.f,0.f,0.f,0.f,0.f};
#pragma unroll
    for (int ci = 0; ci < 16; ++ci) {
      const int c = ci * 4;
      v2f bkv = *(const v2f*)&kcur[lm * KSTRIDE + c + 2 * half];
      L = __builtin_amdgcn_wmma_f32_16x16x4_f32(false, qa[ci], false, bkv,
                                                (short)0, L, false, false);
    }

    // Exact softmax over q per key column.
    // C-layout: lane l holds (q = v + 8*(l/16), k = l%16); column partner = l^16.
    float m = L[0];
#pragma unroll
    for (int v = 1; v < 8; ++v) m = fmaxf(m, L[v]);
    m = fmaxf(m, __shfl_xor(m, 16));
    float e[8];
    float dsum = 0.f;
#pragma unroll
    for (int v = 0; v < 8; ++v) { e[v] = __expf(L[v] - m); dsum += e[v]; }
    dsum += __shfl_xor(dsum, 16);
    const float inv = 1.0f / dsum;  // dsum >= 1
#pragma unroll
    for (int v = 0; v < 8; ++v) { e[v] *= inv; sp[v] += e[v]; }

    // Transpose attn tile through LDS: C-layout -> A-layout (in-order DS per wave)
#pragma unroll
    for (int v = 0; v < 8; ++v) ast[(v + 8 * half) * ASTRIDE + lm] = e[v];
#if !HAVE_ASYNC_LDS
    __syncthreads();
#endif

    // O(16q x 64dv) += attn(16x16) @ Vtile(16x64): 4 dv-tiles x 4 K-chunks
    v2f aa[4];
#pragma unroll
    for (int c2i = 0; c2i < 4; ++c2i) {
      const int c2 = c2i * 4;
      aa[c2i] = *(const v2f*)&ast[lm * ASTRIDE + c2 + 2 * half];
    }
#pragma unroll
    for (int ni = 0; ni < 4; ++ni) {
      const int n0 = ni * 16;
#pragma unroll
      for (int c2i = 0; c2i < 4; ++c2i) {
        const int krow = c2i * 4 + 2 * half;
        v2f bvv;
#if HAVE_ASYNC_LDS
        bvv.x = vcur[(krow    ) * VSTRIDE + n0 + lm];
        bvv.y = vcur[(krow + 1) * VSTRIDE + n0 + lm];
#else
        bvv.x = vtile[(krow    ) * DDIM + n0 + lm];
        bvv.y = vtile[(krow + 1) * DDIM + n0 + lm];
#endif
        acc[ni] = __builtin_amdgcn_wmma_f32_16x16x4_f32(false, aa[c2i], false, bvv,
                                                        (short)0, acc[ni], false, false);
      }
    }
#if HAVE_ASYNC_LDS
    asm volatile("" ::: "memory");
#else
    __syncthreads();
#endif
  }

  // Row-sum reduction: S[q] spread over the 16 lanes of each half
#pragma unroll
  for (int v = 0; v < 8; ++v) {
    float x = sp[v];
    x += __shfl_xor(x, 1);
    x += __shfl_xor(x, 2);
    x += __shfl_xor(x, 4);
    x += __shfl_xor(x, 8);
    sp[v] = x;
  }

  const size_t slot = (size_t)(b * SPLIT + s);
  float* oOut = wsO + slot * (QDIM * DDIM);
#pragma unroll
  for (int ni = 0; ni < 4; ++ni)
#pragma unroll
    for (int v = 0; v < 8; ++v)
      oOut[(v + 8 * half) * DDIM + ni * 16 + lm] = acc[ni][v];

  if (lm == 0) {
    float* sOut = wsS + slot * QDIM;
#pragma unroll
    for (int v = 0; v < 8; ++v) sOut[v + 8 * half] = sp[v];
  }
}

__global__ __launch_bounds__(256)
void inv_attn_finalize(const float* __restrict__ wsO,
                       const float* __restrict__ wsS,
                       float* __restrict__ out) {
  const int i  = blockIdx.x * 256 + threadIdx.x;  // 0..65535
  const int dv = i & 63;
  const int q  = (i >> 6) & 15;
  const int b  = i >> 10;
  float o = 0.f, ssum = 0.f;
#pragma unroll 4
  for (int s = 0; s < SPLIT; ++s) {
    const size_t slot = (size_t)(b * SPLIT + s);
    o    += wsO[slot * (QDIM * DDIM) + q * DDIM + dv];
    ssum += wsS[slot * QDIM + q];
  }
  out[i] = o / (ssum + 1e-8f);
}

extern "C" void kernel_launch(void* const* d_in, const int* in_sizes, int n_in,
                              void* d_out, int out_size, void* d_ws, size_t ws_size,
                              hipStream_t stream) {
  const float* Q = (const float*)d_in[0];
  const float* K = (const float*)d_in[1];
  const float* V = (const float*)d_in[2];
  float* out = (float*)d_out;

  float* wsO = (float*)d_ws;                                   // 64*32*1024 f32
  float* wsS = wsO + (size_t)BATCHES * SPLIT * QDIM * DDIM;    // 64*32*16  f32

  inv_attn_partial<<<dim3(BATCHES * SPLIT), dim3(32), 0, stream>>>(Q, K, V, wsO, wsS);
  inv_attn_finalize<<<dim3((BATCHES * QDIM * DDIM) / 256), dim3(256), 0, stream>>>(wsO, wsS, out);
}